// GreedyRouter_79087527788635
// MI455X (gfx1250) — compile-verified
//
#include <hip/hip_runtime.h>

// MoE greedy router for MI455X (gfx1250).
// d_out layout (floats): [ logits copy | topk_weights | topk_ids | tokens_per_expert ]
//
// Data path (CDNA5-specific):
//   global --(global_load_async_to_lds_b128)--> LDS row --(ds_load_b128)--> compute
//                                               LDS row --(global_store_async_from_lds_b128)--> logits copy out
// ASYNCcnt synchronized with s_wait_asynccnt; double-buffered per wave so the
// DMA of token t+1 and the copy-out of token t overlap the top-k compute on t.

#define N_EXPERTS        256
#define TOP_K            8
#define WAVES_PER_BLOCK  8
#define BLOCK_THREADS    256
#define TOKENS_PER_WAVE  32

typedef __attribute__((address_space(3))) void lds_void_t;

__device__ __forceinline__ unsigned lds_off_u32(void* p) {
    // generic -> AS3 addrspacecast; AS3 pointers are the 32-bit LDS offsets the
    // async-LDS instructions consume in their VDST/VSRC VGPR.
    return (unsigned)(unsigned long long)(lds_void_t*)p;
}

__device__ __forceinline__ void async_load_b128(unsigned lds_off, const void* gaddr) {
    unsigned long long ga = (unsigned long long)gaddr;
    asm volatile("global_load_async_to_lds_b128 %0, %1, off"
                 :: "v"(lds_off), "v"(ga) : "memory");
}

__device__ __forceinline__ void async_store_b128(const void* gaddr, unsigned lds_off) {
    unsigned long long ga = (unsigned long long)gaddr;
    asm volatile("global_store_async_from_lds_b128 %0, %1, off"
                 :: "v"(ga), "v"(lds_off) : "memory");
}

__device__ __forceinline__ void wait_async0() {
    asm volatile("s_wait_asynccnt 0x0" ::: "memory");
}

__global__ void __launch_bounds__(BLOCK_THREADS)
moe_router_kernel(const float* __restrict__ logits,
                  float* __restrict__ out_logits,
                  float* __restrict__ out_w,
                  float* __restrict__ out_ids,
                  float* __restrict__ out_hist,
                  long long n_tokens) {
    __shared__ __align__(16) float rows[2][WAVES_PER_BLOCK][N_EXPERTS];
    __shared__ unsigned hist[N_EXPERTS];

    const int tid = threadIdx.x;
    hist[tid] = 0u;                      // BLOCK_THREADS == N_EXPERTS
    __syncthreads();

    const int lane = tid & 31;
    const int wv   = tid >> 5;
    const long long wave_id = (long long)blockIdx.x * WAVES_PER_BLOCK + wv;
    const long long t0      = wave_id * TOKENS_PER_WAVE;

    const int base0 = 4 * lane;          // expert ids covered by this lane: base0..base0+3
    const int base1 = 128 + 4 * lane;    //                                  base1..base1+3

    if (t0 < n_tokens) {
        // ---- prologue: DMA first row into buffer 0 ----
        {
            const char* g = (const char*)(logits + t0 * N_EXPERTS) + lane * 16;
            unsigned l = lds_off_u32(rows[0][wv]) + (unsigned)(lane * 16);
            async_load_b128(l,       g);
            async_load_b128(l + 512, g + 512);
        }

        for (int i = 0; i < TOKENS_PER_WAVE; ++i) {
            const long long t = t0 + i;
            if (t >= n_tokens) break;
            float* cur = (i & 1) ? rows[1][wv] : rows[0][wv];
            float* nxt = (i & 1) ? rows[0][wv] : rows[1][wv];
            const unsigned cur_off = lds_off_u32(cur) + (unsigned)(lane * 16);

            // Wait: cur's load done (data in LDS) AND the store that previously
            // read nxt has finished (safe to overwrite nxt).
            wait_async0();

            // Pass-through logits copy: LDS -> global, no VGPR staging.
            {
                const char* go = (const char*)(out_logits + t * N_EXPERTS) + lane * 16;
                async_store_b128(go,       cur_off);
                async_store_b128(go + 512, cur_off + 512);
            }
            // Prefetch next row into the other buffer (overlaps compute below).
            if (i + 1 < TOKENS_PER_WAVE && t + 1 < n_tokens) {
                const char* g = (const char*)(logits + (t + 1) * N_EXPERTS) + lane * 16;
                unsigned l = lds_off_u32(nxt) + (unsigned)(lane * 16);
                async_load_b128(l,       g);
                async_load_b128(l + 512, g + 512);
            }

            // ---- compute: read row from LDS (ds_load_b128 x2 per lane) ----
            const float4* cf = (const float4*)cur;
            float4 ra = cf[lane];
            float4 rb = cf[lane + 32];
            float v[8] = {ra.x, ra.y, ra.z, ra.w, rb.x, rb.y, rb.z, rb.w};

            float wk[TOP_K]; int ik[TOP_K];
#pragma unroll
            for (int k = 0; k < TOP_K; ++k) {
                // lane-local argmax, lowest index wins ties (ascending scan, strict >)
                float bv = v[0]; int bi = base0;
#pragma unroll
                for (int j = 1; j < 8; ++j) {
                    const int idx = (j < 4) ? (base0 + j) : (base1 + j - 4);
                    if (v[j] > bv) { bv = v[j]; bi = idx; }
                }
                // value-only wave32 butterfly max (5 swizzles)
                float rv = bv;
#pragma unroll
                for (int off = 16; off >= 1; off >>= 1)
                    rv = fmaxf(rv, __shfl_xor(rv, off, 32));
                // winner index: lowest candidate lane, then its local lowest index
                const unsigned mask = __builtin_amdgcn_ballot_w32(bv == rv);
                const int src = (int)__builtin_ctz(mask);
                const int ri  = __builtin_amdgcn_readlane(bi, src);
                wk[k] = rv; ik[k] = ri;
                // branchless clear of the consumed slot (only owner lane matches)
#pragma unroll
                for (int j = 0; j < 8; ++j) {
                    const int idx = (j < 4) ? (base0 + j) : (base1 + j - 4);
                    if (idx == ri) v[j] = -__builtin_inff();
                }
            }

            // renormalized top-k softmax: exp(l - max) / sum over the selected 8
            const float m = wk[0];
            float e[TOP_K]; float s = 0.f;
#pragma unroll
            for (int k = 0; k < TOP_K; ++k) { e[k] = __expf(wk[k] - m); s += e[k]; }
            const float inv = 1.0f / s;

            if (lane == 0) {
                float4* wp = (float4*)(out_w + t * TOP_K);
                wp[0] = make_float4(e[0] * inv, e[1] * inv, e[2] * inv, e[3] * inv);
                wp[1] = make_float4(e[4] * inv, e[5] * inv, e[6] * inv, e[7] * inv);
                float4* ip = (float4*)(out_ids + t * TOP_K);
                ip[0] = make_float4((float)ik[0], (float)ik[1], (float)ik[2], (float)ik[3]);
                ip[1] = make_float4((float)ik[4], (float)ik[5], (float)ik[6], (float)ik[7]);
#pragma unroll
                for (int k = 0; k < TOP_K; ++k) atomicAdd(&hist[ik[k]], 1u);
            }
        }
    }

    wait_async0();          // last copy-out store has consumed LDS
    __syncthreads();
    atomicAdd(&out_hist[tid], (float)hist[tid]);   // per-block histogram flush
}

__global__ void init_hist_kernel(float* __restrict__ out_hist) {
    out_hist[threadIdx.x] = 0.0f;
}

extern "C" void kernel_launch(void* const* d_in, const int* in_sizes, int n_in,
                              void* d_out, int out_size, void* d_ws, size_t ws_size,
                              hipStream_t stream) {
    const float* logits = (const float*)d_in[0];
    const long long n_tokens = (long long)in_sizes[0] / N_EXPERTS;

    float* out        = (float*)d_out;
    float* out_logits = out;
    float* out_w      = out_logits + n_tokens * N_EXPERTS;
    float* out_ids    = out_w + n_tokens * TOP_K;
    float* out_hist   = out_ids + n_tokens * TOP_K;

    init_hist_kernel<<<1, N_EXPERTS, 0, stream>>>(out_hist);

    const long long tokens_per_block = (long long)WAVES_PER_BLOCK * TOKENS_PER_WAVE;
    const int blocks = (int)((n_tokens + tokens_per_block - 1) / tokens_per_block);
    moe_router_kernel<<<blocks, BLOCK_THREADS, 0, stream>>>(
        logits, out_logits, out_w, out_ids, out_hist, n_tokens);
}